// LaplacianAttention_59777354825748
// MI455X (gfx1250) — compile-verified
//
#include <hip/hip_runtime.h>
#include <math.h>

typedef __attribute__((ext_vector_type(2))) float v2f;
typedef __attribute__((ext_vector_type(4))) float v4f;
typedef __attribute__((ext_vector_type(8))) float v8f;

#define HWPLANE 16384   // 128*128
#define CCH 256
#define BB 16
#define H3 48           // 3 * C/r
#define NPLANES 4096    // B*C

// ---------------------------------------------------------------------------
// Kernel 1: per-(b,c) plane global avg + max.  4096 blocks x 256 threads.
// Each block reduces one contiguous 16384-float plane (64 KB).
// ---------------------------------------------------------------------------
__global__ void la_pool_kernel(const float* __restrict__ x,
                               float* __restrict__ avg_out,
                               float* __restrict__ max_out) {
    const int plane = blockIdx.x;
    const v4f* xv = reinterpret_cast<const v4f*>(x + (size_t)plane * HWPLANE);
    const int t = threadIdx.x;

    float s = 0.0f;
    float m = -3.402823466e38f;
#pragma unroll
    for (int i = 0; i < 16; ++i) {
        v4f v = xv[t + i * 256];
        s += v.x + v.y + v.z + v.w;
        m = fmaxf(m, fmaxf(fmaxf(v.x, v.y), fmaxf(v.z, v.w)));
    }
    // wave32 butterfly reduction
#pragma unroll
    for (int off = 16; off >= 1; off >>= 1) {
        s += __shfl_xor(s, off, 32);
        m = fmaxf(m, __shfl_xor(m, off, 32));
    }
    __shared__ float ls[8];
    __shared__ float lm[8];
    const int wid = t >> 5;
    if ((t & 31) == 0) { ls[wid] = s; lm[wid] = m; }
    __syncthreads();
    if (t == 0) {
        float ts = 0.0f, tm = -3.402823466e38f;
#pragma unroll
        for (int w = 0; w < 8; ++w) { ts += ls[w]; tm = fmaxf(tm, lm[w]); }
        avg_out[plane] = ts * (1.0f / (float)HWPLANE);
        max_out[plane] = tm;
    }
}

// ---------------------------------------------------------------------------
// Kernel 2: the whole MLP + gating on one wave32 using V_WMMA_F32_16X16X4_F32.
// pools = [avg(16x256) | max(16x256)] ; gate = (16x256) written to ws.
// ---------------------------------------------------------------------------
__device__ __forceinline__ float la_sigmoid(float v) {
    return 1.0f / (1.0f + __expf(-v));
}

__global__ __launch_bounds__(32) void la_gate_kernel(
        const float* __restrict__ pools,
        const float* __restrict__ w1, const float* __restrict__ b1,
        const float* __restrict__ w2, const float* __restrict__ b2,
        const float* __restrict__ w3, const float* __restrict__ b3,
        const float* __restrict__ w4, const float* __restrict__ b4,
        float* __restrict__ gate) {
    // h for both branches: [branch][16 rows][48 cols]
    __shared__ float hbuf[2][BB * H3];

    const int lane = threadIdx.x;            // 0..31, wave32, EXEC all ones
    const bool lo  = lane < 16;
    const int  mn  = lane & 15;              // M row (A) / N col (B) index

    const float* wtab[3] = { w1, w2, w3 };
    const float* btab[3] = { b1, b2, b3 };

    // ---- first layer: h_wi = relu(p @ Wc_wi^T + b_wi), Wc stride 9, tap 4 ----
    for (int br = 0; br < 2; ++br) {
        const float* p = pools + br * (BB * CCH);
#pragma unroll
        for (int wi = 0; wi < 3; ++wi) {
            const float* wc = wtab[wi];
            v8f acc = {};
            for (int k0 = 0; k0 < CCH; k0 += 4) {
                const int kA = k0 + (lo ? 0 : 2);
                // A[m][k] = p[m][k]  (16x4 tile, lanes<16: K=k0,k0+1; lanes>=16: K=k0+2,k0+3)
                v2f a = *reinterpret_cast<const v2f*>(p + mn * CCH + kA);
                // B[k][n] = Wc[n][k] = w[n*256*9 + k*9 + 4]
                v2f b;
                b.x = wc[mn * (CCH * 9) + kA * 9 + 4];
                b.y = wc[mn * (CCH * 9) + (kA + 1) * 9 + 4];
                acc = __builtin_amdgcn_wmma_f32_16x16x4_f32(
                        false, a, false, b, (short)0, acc, false, false);
            }
            const float bias = btab[wi][mn];
#pragma unroll
            for (int r = 0; r < 8; ++r) {
                const int M = r + (lo ? 0 : 8);  // C/D layout: lanes>=16 hold M=r+8
                hbuf[br][M * H3 + wi * 16 + mn] = fmaxf(acc[r] + bias, 0.0f);
            }
        }
    }
    __syncthreads();

    // ---- second layer: z = h @ W4c^T + b4 ; gate = sig(sig(z_avg)+sig(z_max)) ----
    for (int t = 0; t < 16; ++t) {
        const int n0   = t * 16;
        const int ncol = n0 + mn;
        v8f accA = {};
        v8f accM = {};
#pragma unroll
        for (int j0 = 0; j0 < H3; j0 += 4) {
            const int jA = j0 + (lo ? 0 : 2);
            v2f b;  // shared between both branches
            b.x = w4[ncol * (H3 * 9) + jA * 9 + 4];
            b.y = w4[ncol * (H3 * 9) + (jA + 1) * 9 + 4];
            v2f aA = *reinterpret_cast<const v2f*>(&hbuf[0][mn * H3 + jA]);
            v2f aM = *reinterpret_cast<const v2f*>(&hbuf[1][mn * H3 + jA]);
            accA = __builtin_amdgcn_wmma_f32_16x16x4_f32(
                     false, aA, false, b, (short)0, accA, false, false);
            accM = __builtin_amdgcn_wmma_f32_16x16x4_f32(
                     false, aM, false, b, (short)0, accM, false, false);
        }
        const float bias = b4[ncol];
#pragma unroll
        for (int r = 0; r < 8; ++r) {
            const int M = r + (lo ? 0 : 8);
            const float g = la_sigmoid(la_sigmoid(accA[r] + bias) +
                                       la_sigmoid(accM[r] + bias));
            gate[M * CCH + ncol] = g;
        }
    }
}

// ---------------------------------------------------------------------------
// Kernel 3: y = x * gate[b][c].  Planes walked in REVERSE so the second read
// of x starts on the ~192 MB that kernel 1 left resident in L2.  Output is
// never re-read -> non-temporal stores; x is last-use -> non-temporal loads.
// ---------------------------------------------------------------------------
__global__ void la_scale_kernel(const float* __restrict__ x,
                                const float* __restrict__ gate,
                                float* __restrict__ out) {
    const int plane = (int)(NPLANES - 1) - (int)blockIdx.x;  // reverse order
    const float g = gate[plane];
    const size_t base = (size_t)plane * HWPLANE;
    const v4f* xv = reinterpret_cast<const v4f*>(x + base);
    v4f* ov = reinterpret_cast<v4f*>(out + base);
    const int t = threadIdx.x;
#pragma unroll
    for (int i = 0; i < 16; ++i) {
        v4f v = __builtin_nontemporal_load(&xv[t + i * 256]);
        v4f r;
        r.x = v.x * g; r.y = v.y * g; r.z = v.z * g; r.w = v.w * g;
        __builtin_nontemporal_store(r, &ov[t + i * 256]);
    }
}

// ---------------------------------------------------------------------------
extern "C" void kernel_launch(void* const* d_in, const int* in_sizes, int n_in,
                              void* d_out, int out_size, void* d_ws, size_t ws_size,
                              hipStream_t stream) {
    const float* x  = (const float*)d_in[0];
    const float* w1 = (const float*)d_in[1];
    const float* b1 = (const float*)d_in[2];
    const float* w2 = (const float*)d_in[3];
    const float* b2 = (const float*)d_in[4];
    const float* w3 = (const float*)d_in[5];
    const float* b3 = (const float*)d_in[6];
    const float* w4 = (const float*)d_in[7];
    const float* b4 = (const float*)d_in[8];
    float* out = (float*)d_out;

    float* ws   = (float*)d_ws;
    float* avg  = ws;                 // 4096 floats
    float* mx   = ws + NPLANES;       // 4096 floats
    float* gate = ws + 2 * NPLANES;   // 4096 floats

    la_pool_kernel<<<NPLANES, 256, 0, stream>>>(x, avg, mx);
    la_gate_kernel<<<1, 32, 0, stream>>>(avg, w1, b1, w2, b2, w3, b3, w4, b4, gate);
    la_scale_kernel<<<NPLANES, 256, 0, stream>>>(x, gate, out);
}